// LSTM101_10342281249033
// MI455X (gfx1250) — compile-verified
//
#include <hip/hip_runtime.h>
#include <hip/hip_bf16.h>

#define HID 20
#define G4  80   // 4*HID
#define BLK 8    // xg1 register-prefetch ring depth (steps)

typedef __attribute__((ext_vector_type(2))) float v2f;
typedef __attribute__((ext_vector_type(8))) float v8f;

// ---------------------------------------------------------------------------
// Fast activations on the recurrent critical path: v_exp_f32 + v_rcp_f32.
// tanh(x) = 1 - 2/(e^{2x}+1): saturates to +/-1 correctly at |x| large.
// ---------------------------------------------------------------------------
__device__ __forceinline__ float frcp(float v)  { return __builtin_amdgcn_rcpf(v); }
__device__ __forceinline__ float sigf(float v)  { return frcp(1.0f + __expf(-v)); }
__device__ __forceinline__ float tanhfast(float v) {
    float e = __expf(2.0f * v);
    return 1.0f - 2.0f * frcp(e + 1.0f);
}

// ---------------------------------------------------------------------------
// Kernel A: xg1[t][g] = b1[g] + sum_d x[t][d] * w_ih1[g][d]
// Parallel over time via V_WMMA_F32_16X16X4_F32 (f32-exact).
// K padded 9->12 in LDS (zero fill), so all fragment loads are straight-line
// ds_loads and EXEC stays all-ones through the WMMA region.
// One wave = 16 time rows; 3 K-chunks x 5 N-tiles = 15 WMMAs per wave.
// ---------------------------------------------------------------------------
__global__ void __launch_bounds__(256) xg1_wmma_kernel(
    const float* __restrict__ x,      // [T,9]
    const float* __restrict__ w_ih1,  // [80,9]
    const float* __restrict__ b1,     // [80]
    float* __restrict__ xg1,          // [T,80]
    int T)
{
    __shared__ float sw[G4 * 12];     // weights padded [80][12]
    __shared__ float sb[G4];
    __shared__ float sx[8][16 * 12];  // per-wave x tile padded [16][12]

    const int tid  = threadIdx.x;
    const int lane = tid & 31;
    const int wave = tid >> 5;
    const int t0   = (blockIdx.x * 8 + wave) * 16;

    // Stage zero-padded weights + bias (whole block cooperates)
    for (int idx = tid; idx < G4 * 12; idx += 256) {
        const int g = idx / 12, k = idx % 12;
        sw[idx] = (k < 9) ? w_ih1[g * 9 + k] : 0.0f;
    }
    if (tid < G4) sb[tid] = b1[tid];

    // Stage this wave's zero-padded x tile
    for (int idx = lane; idx < 16 * 12; idx += 32) {
        const int r = idx / 12, k = idx % 12;
        sx[wave][idx] = (k < 9) ? x[(size_t)(t0 + r) * 9 + k] : 0.0f;
    }
    __syncthreads();

    const int mrow  = lane & 15;        // M (A) / N (B,D) index for this lane
    const int khalf = (lane >> 4) * 2;  // 0 for lanes 0-15, 2 for lanes 16-31

    // A fragments: VGPR0 = K=k0+khalf, VGPR1 = K=k0+khalf+1
    v2f a[3];
#pragma unroll
    for (int c = 0; c < 3; ++c) {
        const int k = c * 4 + khalf;
        a[c].x = sx[wave][mrow * 12 + k];
        a[c].y = sx[wave][mrow * 12 + k + 1];
    }

#pragma unroll
    for (int n0 = 0; n0 < G4; n0 += 16) {
        const int g = n0 + mrow;
        const float bias = sb[g];
        v8f acc = { bias, bias, bias, bias, bias, bias, bias, bias };

#pragma unroll
        for (int c = 0; c < 3; ++c) {
            const int k = c * 4 + khalf;
            v2f b;
            b.x = sw[g * 12 + k];
            b.y = sw[g * 12 + k + 1];
            acc = __builtin_amdgcn_wmma_f32_16x16x4_f32(
                false, a[c], false, b, (short)0, acc, false, false);
        }

        // D layout: VGPR r -> M = r (+8 for lanes 16-31), N = lane&15
        const int trow = t0 + ((lane >> 4) ? 8 : 0);
#pragma unroll
        for (int r = 0; r < 8; ++r) {
            xg1[(size_t)(trow + r) * G4 + g] = acc[r];
        }
    }
}

// ---------------------------------------------------------------------------
// Sequential recurrence, single block, layers pipelined one step apart.
//   tid   0..79  : layer-1 gate j = tid        (step i)   -- applies activation
//   tid  80..159 : layer-2 gate j = tid-80     (step i-1) -- applies activation
//   tid   0..19  : layer-1 cell/h update
//   tid  20..39  : layer-2 cell/h update
//   tid  160     : projection y[i-2]
// Gate order (PyTorch): [0,20)=i  [20,40)=f  [40,60)=g  [60,80)=o
// xg1 is streamed through a BLK-deep register ring (loads issued one block
// ahead), so L2 latency never sits on the recurrent chain.
// ---------------------------------------------------------------------------
__global__ void __launch_bounds__(192) lstm_seq_kernel(
    const float* __restrict__ xg1,   // [T,80] precomputed input projection
    const float* __restrict__ w_hh1, // [80,20]
    const float* __restrict__ w_ih2, // [80,20]
    const float* __restrict__ w_hh2, // [80,20]
    const float* __restrict__ b2,    // [80]
    const float* __restrict__ w_p,   // [20]
    const float* __restrict__ b_p,   // [1]
    float* __restrict__ y,           // [T]
    int T)
{
    __shared__ __align__(16) float h1[HID];
    __shared__ __align__(16) float h2[HID];
    __shared__ float g1[G4], g2[G4];   // ACTIVATED gates

    const int tid = threadIdx.x;

    // Per-thread weight registers (fully unrolled dots -> stays in VGPRs)
    float wr1[HID];
    float wi2[HID], wr2[HID];
    float wp[HID];
    float bias2 = 0.0f, bp = 0.0f;

    if (tid < G4) {
#pragma unroll
        for (int k = 0; k < HID; ++k) wr1[k] = w_hh1[tid * HID + k];
    } else if (tid < 2 * G4) {
        const int j = tid - G4;
#pragma unroll
        for (int k = 0; k < HID; ++k) {
            wi2[k] = w_ih2[j * HID + k];
            wr2[k] = w_hh2[j * HID + k];
        }
        bias2 = b2[j];
    } else if (tid == 2 * G4) {
#pragma unroll
        for (int k = 0; k < HID; ++k) wp[k] = w_p[k];
        bp = b_p[0];
    }

    float c1 = 0.0f, c2 = 0.0f;
    if (tid < HID) { h1[tid] = 0.0f; h2[tid] = 0.0f; }

    // Preload xg1 ring for steps 0..BLK-1
    float xb[BLK], xn[BLK];
    if (tid < G4) {
#pragma unroll
        for (int u = 0; u < BLK; ++u)
            xb[u] = (u < T) ? xg1[(size_t)u * G4 + tid] : 0.0f;
    }
    __syncthreads();

    const float4* h1v = (const float4*)h1;
    const float4* h2v = (const float4*)h2;

    for (int i0 = 0; i0 < T + 2; i0 += BLK) {
        // Issue next block's xg1 loads; they have ~BLK step-latencies to land.
        if (tid < G4) {
#pragma unroll
            for (int u = 0; u < BLK; ++u) {
                const int ip = i0 + BLK + u;
                xn[u] = (ip < T) ? xg1[(size_t)ip * G4 + tid] : 0.0f;
            }
        }

#pragma unroll
        for (int u = 0; u < BLK; ++u) {
            const int i = i0 + u;

            // ------------- phase A: gate matvecs + activation + projection ----
            if (tid < G4) {
                if (i < T) {
                    float acc = 0.0f;
#pragma unroll
                    for (int q = 0; q < 5; ++q) {
                        const float4 hv = h1v[q];
                        acc = fmaf(hv.x, wr1[4 * q + 0], acc);
                        acc = fmaf(hv.y, wr1[4 * q + 1], acc);
                        acc = fmaf(hv.z, wr1[4 * q + 2], acc);
                        acc = fmaf(hv.w, wr1[4 * q + 3], acc);
                    }
                    acc += xb[u];  // prefetched; off the dot's dependency chain
                    g1[tid] = (tid < 2 * HID || tid >= 3 * HID) ? sigf(acc)
                                                                : tanhfast(acc);
                }
            } else if (tid < 2 * G4) {
                if (i >= 1 && i <= T) {   // layer-2 step i-1: uses h1[i-1], h2[i-2]
                    const int j = tid - G4;
                    float acc = bias2;
#pragma unroll
                    for (int q = 0; q < 5; ++q) {
                        const float4 ha = h1v[q];
                        acc = fmaf(ha.x, wi2[4 * q + 0], acc);
                        acc = fmaf(ha.y, wi2[4 * q + 1], acc);
                        acc = fmaf(ha.z, wi2[4 * q + 2], acc);
                        acc = fmaf(ha.w, wi2[4 * q + 3], acc);
                    }
#pragma unroll
                    for (int q = 0; q < 5; ++q) {
                        const float4 hb = h2v[q];
                        acc = fmaf(hb.x, wr2[4 * q + 0], acc);
                        acc = fmaf(hb.y, wr2[4 * q + 1], acc);
                        acc = fmaf(hb.z, wr2[4 * q + 2], acc);
                        acc = fmaf(hb.w, wr2[4 * q + 3], acc);
                    }
                    g2[j] = (j < 2 * HID || j >= 3 * HID) ? sigf(acc)
                                                          : tanhfast(acc);
                }
            } else if (tid == 2 * G4) {
                if (i >= 2 && i <= T + 1) {      // h2 currently holds step i-2
                    float acc = bp;
#pragma unroll
                    for (int q = 0; q < 5; ++q) {
                        const float4 hb = h2v[q];
                        acc = fmaf(hb.x, wp[4 * q + 0], acc);
                        acc = fmaf(hb.y, wp[4 * q + 1], acc);
                        acc = fmaf(hb.z, wp[4 * q + 2], acc);
                        acc = fmaf(hb.w, wp[4 * q + 3], acc);
                    }
                    y[i - 2] = acc;
                }
            }
            __syncthreads();

            // ------------- phase B: cell/hidden updates (gates pre-activated) -
            if (tid < HID) {
                if (i < T) {
                    c1 = fmaf(g1[HID + tid], c1, g1[tid] * g1[2 * HID + tid]);
                    h1[tid] = g1[3 * HID + tid] * tanhfast(c1);
                }
            } else if (tid < 2 * HID) {
                const int v = tid - HID;
                if (i >= 1 && i <= T) {
                    c2 = fmaf(g2[HID + v], c2, g2[v] * g2[2 * HID + v]);
                    h2[v] = g2[3 * HID + v] * tanhfast(c2);
                }
            }
            __syncthreads();
        }

        if (tid < G4) {
#pragma unroll
            for (int u = 0; u < BLK; ++u) xb[u] = xn[u];
        }
    }
}

// ---------------------------------------------------------------------------
// Fallback (workspace too small): input projection fused into the layer-1
// gate threads (9 extra FMAs from L0-broadcast x rows). Same pipeline.
// ---------------------------------------------------------------------------
__global__ void __launch_bounds__(192) lstm_seq_fused_kernel(
    const float* __restrict__ x,     // [T,9]
    const float* __restrict__ w_ih1, // [80,9]
    const float* __restrict__ b1,    // [80]
    const float* __restrict__ w_hh1,
    const float* __restrict__ w_ih2,
    const float* __restrict__ w_hh2,
    const float* __restrict__ b2,
    const float* __restrict__ w_p,
    const float* __restrict__ b_p,
    float* __restrict__ y,
    int T)
{
    __shared__ __align__(16) float h1[HID];
    __shared__ __align__(16) float h2[HID];
    __shared__ float g1[G4], g2[G4];

    const int tid = threadIdx.x;

    float wi1[9];
    float wr1[HID];
    float wi2[HID], wr2[HID];
    float wp[HID];
    float bias1 = 0.0f, bias2 = 0.0f, bp = 0.0f;

    if (tid < G4) {
#pragma unroll
        for (int d = 0; d < 9; ++d) wi1[d] = w_ih1[tid * 9 + d];
#pragma unroll
        for (int k = 0; k < HID; ++k) wr1[k] = w_hh1[tid * HID + k];
        bias1 = b1[tid];
    } else if (tid < 2 * G4) {
        const int j = tid - G4;
#pragma unroll
        for (int k = 0; k < HID; ++k) {
            wi2[k] = w_ih2[j * HID + k];
            wr2[k] = w_hh2[j * HID + k];
        }
        bias2 = b2[j];
    } else if (tid == 2 * G4) {
#pragma unroll
        for (int k = 0; k < HID; ++k) wp[k] = w_p[k];
        bp = b_p[0];
    }

    float c1 = 0.0f, c2 = 0.0f;
    if (tid < HID) { h1[tid] = 0.0f; h2[tid] = 0.0f; }
    __syncthreads();

    const float4* h1v = (const float4*)h1;
    const float4* h2v = (const float4*)h2;

    for (int i = 0; i < T + 2; ++i) {
        if (tid < G4) {
            if (i < T) {
                float xa = bias1;
                const float* xr = x + (size_t)i * 9;
#pragma unroll
                for (int d = 0; d < 9; ++d) xa = fmaf(xr[d], wi1[d], xa);
                float acc = 0.0f;
#pragma unroll
                for (int q = 0; q < 5; ++q) {
                    const float4 hv = h1v[q];
                    acc = fmaf(hv.x, wr1[4 * q + 0], acc);
                    acc = fmaf(hv.y, wr1[4 * q + 1], acc);
                    acc = fmaf(hv.z, wr1[4 * q + 2], acc);
                    acc = fmaf(hv.w, wr1[4 * q + 3], acc);
                }
                acc += xa;
                g1[tid] = (tid < 2 * HID || tid >= 3 * HID) ? sigf(acc)
                                                            : tanhfast(acc);
            }
        } else if (tid < 2 * G4) {
            if (i >= 1 && i <= T) {
                const int j = tid - G4;
                float acc = bias2;
#pragma unroll
                for (int q = 0; q < 5; ++q) {
                    const float4 ha = h1v[q];
                    acc = fmaf(ha.x, wi2[4 * q + 0], acc);
                    acc = fmaf(ha.y, wi2[4 * q + 1], acc);
                    acc = fmaf(ha.z, wi2[4 * q + 2], acc);
                    acc = fmaf(ha.w, wi2[4 * q + 3], acc);
                }
#pragma unroll
                for (int q = 0; q < 5; ++q) {
                    const float4 hb = h2v[q];
                    acc = fmaf(hb.x, wr2[4 * q + 0], acc);
                    acc = fmaf(hb.y, wr2[4 * q + 1], acc);
                    acc = fmaf(hb.z, wr2[4 * q + 2], acc);
                    acc = fmaf(hb.w, wr2[4 * q + 3], acc);
                }
                g2[j] = (j < 2 * HID || j >= 3 * HID) ? sigf(acc)
                                                      : tanhfast(acc);
            }
        } else if (tid == 2 * G4) {
            if (i >= 2 && i <= T + 1) {
                float acc = bp;
#pragma unroll
                for (int q = 0; q < 5; ++q) {
                    const float4 hb = h2v[q];
                    acc = fmaf(hb.x, wp[4 * q + 0], acc);
                    acc = fmaf(hb.y, wp[4 * q + 1], acc);
                    acc = fmaf(hb.z, wp[4 * q + 2], acc);
                    acc = fmaf(hb.w, wp[4 * q + 3], acc);
                }
                y[i - 2] = acc;
            }
        }
        __syncthreads();

        if (tid < HID) {
            if (i < T) {
                c1 = fmaf(g1[HID + tid], c1, g1[tid] * g1[2 * HID + tid]);
                h1[tid] = g1[3 * HID + tid] * tanhfast(c1);
            }
        } else if (tid < 2 * HID) {
            const int v = tid - HID;
            if (i >= 1 && i <= T) {
                c2 = fmaf(g2[HID + v], c2, g2[v] * g2[2 * HID + v]);
                h2[v] = g2[3 * HID + v] * tanhfast(c2);
            }
        }
        __syncthreads();
    }
}

extern "C" void kernel_launch(void* const* d_in, const int* in_sizes, int n_in,
                              void* d_out, int out_size, void* d_ws, size_t ws_size,
                              hipStream_t stream) {
    const float* x    = (const float*)d_in[0];
    const float* wih1 = (const float*)d_in[1];
    const float* whh1 = (const float*)d_in[2];
    const float* b1   = (const float*)d_in[3];
    const float* wih2 = (const float*)d_in[4];
    const float* whh2 = (const float*)d_in[5];
    const float* b2   = (const float*)d_in[6];
    const float* wp   = (const float*)d_in[7];
    const float* bp   = (const float*)d_in[8];
    float* y = (float*)d_out;

    const int T = in_sizes[0] / 9;                 // 262144
    const size_t need = (size_t)T * G4 * sizeof(float);

    if (ws_size >= need && (T % 128) == 0) {
        float* xg1 = (float*)d_ws;
        const int tiles  = T / 16;                 // 16 time rows per wave
        const int blocks = tiles / 8;              // 8 waves per block
        xg1_wmma_kernel<<<blocks, 256, 0, stream>>>(x, wih1, b1, xg1, T);
        lstm_seq_kernel<<<1, 192, 0, stream>>>(xg1, whh1, wih2, whh2, b2, wp, bp, y, T);
    } else {
        lstm_seq_fused_kernel<<<1, 192, 0, stream>>>(x, wih1, b1, whh1, wih2, whh2,
                                                     b2, wp, bp, y, T);
    }
}